// NodeAttention_1425929142443
// MI455X (gfx1250) — compile-verified
//
#include <hip/hip_runtime.h>

// GAT-style node attention, fused for MI455X (gfx1250, wave32, WMMA).
// B=16384, N=32, D=256, C=128, H=8.

typedef __attribute__((ext_vector_type(2))) float v2f;
typedef __attribute__((ext_vector_type(8))) float v8f;

#define B_TOT   16384
#define N_NB    32
#define D_IN    256
#define C_OUT   128
#define H_HEADS 8
#define ALPHA   0.2f

#define NB_PER_BLK 2          // nodes per block -> 64 neighbor rows
#define THREADS    64         // 2 wave32 waves, each owns a 32x128 output tile
#define LDS_S      133        // padded row stride (133 mod 64 = 5 -> conflict-free)

// ---------------------------------------------------------------------------
// Prep: Wa_src[d][h] = sum_c W[d][c] * a_src[h][c]   (256x8, 8 KB)
//       Wt[c][d]     = W[d][c]                        (128x256, 128 KB)
// Wt makes every WMMA B-fragment a contiguous 8-byte load.
// ---------------------------------------------------------------------------
__global__ __launch_bounds__(256) void gat_prep(
    const float* __restrict__ W,       // (D, C)
    const float* __restrict__ attnk,   // (H, 2C)
    float* __restrict__ wa_src,        // (D, H)
    float* __restrict__ wt)            // (C, D)
{
    const int t = threadIdx.x;
    for (int i = t; i < D_IN * H_HEADS; i += 256) {
        const int d = i >> 3, h = i & 7;
        float s = 0.f;
        for (int c = 0; c < C_OUT; ++c)
            s += W[d * C_OUT + c] * attnk[h * (2 * C_OUT) + c];
        wa_src[i] = s;
    }
    for (int i = t; i < D_IN * C_OUT; i += 256) {
        const int c = i >> 8, d = i & 255;
        wt[i] = W[d * C_OUT + c];
    }
}

// ---------------------------------------------------------------------------
// Fused kernel, 2 nodes (64 neighbor rows) per 64-thread block.
//   1) Each wave computes a 32x128 slice of neigh_p with f32 WMMA:
//      two 16-row A tiles share every B fragment -> 16 WMMAs per 10 loads.
//   2) scores[b][h][n] = node_src_score + <neigh_p[b,n,:], a_dst[h,:]>
//   3) leaky-relu, softmax over n
//   4) z[b][h][c] = sum_n coef * neigh_p[b,n,c]  (+bias), coalesced store.
// ---------------------------------------------------------------------------
__global__ __launch_bounds__(THREADS) void gat_fused(
    const float* __restrict__ node_feats,   // (B, D)
    const float* __restrict__ neigh,        // (B, N, D)
    const float* __restrict__ attnk,        // (H, 2C)
    const float* __restrict__ bias,         // (H*C)
    const float* __restrict__ wa_src,       // (D, H)
    const float* __restrict__ wt,           // (C, D)
    float* __restrict__ out)                // (B, H*C)
{
    __shared__ float lds_p[NB_PER_BLK * N_NB * LDS_S];        // 64 x 133 fp32
    __shared__ float lds_sc[NB_PER_BLK * H_HEADS * N_NB];     // 512
    __shared__ float lds_ssrc[NB_PER_BLK * H_HEADS];          // 16

    const int t     = threadIdx.x;
    const int lane  = t & 31;
    const int wv    = t >> 5;                    // 0..1, owns rows [32*wv, 32*wv+32)
    const int bBase = blockIdx.x * NB_PER_BLK;

    // ---- WMMA f32 GEMM: 32x128 per wave, K = 256 in steps of 4 ----
    // A 16x4 f32 layout: lane l holds (M = l&15, K = klo, klo+1), klo=(l>>4)*2.
    // B 4x16 f32 layout: lane l holds (K = klo, klo+1, N = l&15).
    const int  col15 = lane & 15;
    const int  klo   = (lane >> 4) * 2;
    const long gRow0 = (long)bBase * N_NB + wv * 32 + col15;
    const float* aPtr0 = neigh + gRow0 * D_IN + klo;
    const float* aPtr1 = aPtr0 + 16 * D_IN;

    const v8f vzero = {0.f, 0.f, 0.f, 0.f, 0.f, 0.f, 0.f, 0.f};
    v8f acc0[8], acc1[8];
#pragma unroll
    for (int ct = 0; ct < 8; ++ct) { acc0[ct] = vzero; acc1[ct] = vzero; }

    for (int kk = 0; kk < D_IN / 4; ++kk) {
        const v2f a0 = *(const v2f*)(aPtr0 + kk * 4);
        const v2f a1 = *(const v2f*)(aPtr1 + kk * 4);
#pragma unroll
        for (int ct = 0; ct < 8; ++ct) {
            const v2f b = *(const v2f*)(wt + (ct * 16 + col15) * D_IN + kk * 4 + klo);
            acc0[ct] = __builtin_amdgcn_wmma_f32_16x16x4_f32(
                false, a0, false, b, (short)0, acc0[ct], false, false);
            acc1[ct] = __builtin_amdgcn_wmma_f32_16x16x4_f32(
                false, a1, false, b, (short)0, acc1[ct], false, false);
        }
    }

    // Spill C/D tiles to LDS: lane l, reg v -> row rbase + v, col ct*16 + (l&15).
    {
        const int rbase0 = wv * 32 + (lane >> 4) * 8;
        const int rbase1 = rbase0 + 16;
#pragma unroll
        for (int ct = 0; ct < 8; ++ct)
#pragma unroll
            for (int v = 0; v < 8; ++v) {
                lds_p[(rbase0 + v) * LDS_S + ct * 16 + col15] = acc0[ct][v];
                lds_p[(rbase1 + v) * LDS_S + ct * 16 + col15] = acc1[ct][v];
            }
    }

    // Node source scores: s_src[b][h] = <node_feats[b,:], Wa_src[:,h]>
    if (t < NB_PER_BLK * H_HEADS) {
        const int b = t >> 3, h = t & 7;
        const float* nf = node_feats + (long)(bBase + b) * D_IN;
        float s = 0.f;
        for (int d = 0; d < D_IN; ++d) s += nf[d] * wa_src[d * H_HEADS + h];
        lds_ssrc[t] = s;
    }
    __syncthreads();

    // Scores + leaky-relu: 512 scores, 8 per thread.
    for (int j = 0; j < 8; ++j) {
        const int i = t + j * THREADS;
        const int b = i >> 8, h = (i >> 5) & 7, n = i & 31;
        const float* row = lds_p + (b * N_NB + n) * LDS_S;
        const float* ad  = attnk + h * (2 * C_OUT) + C_OUT;   // a_dst[h,:]
        float s = lds_ssrc[b * H_HEADS + h];
        for (int c = 0; c < C_OUT; ++c) s += row[c] * ad[c];
        s = (s > 0.f) ? s : ALPHA * s;
        lds_sc[(b * H_HEADS + h) * N_NB + n] = s;
    }
    __syncthreads();

    // Softmax over n=32 for each of the 16 (b,h) pairs.
    if (t < NB_PER_BLK * H_HEADS) {
        float* sc = lds_sc + t * N_NB;
        float m = sc[0];
        for (int n = 1; n < N_NB; ++n) m = fmaxf(m, sc[n]);
        float sum = 0.f;
        for (int n = 0; n < N_NB; ++n) sum += __expf(sc[n] - m);
        const float inv = 1.f / sum;
        for (int n = 0; n < N_NB; ++n) sc[n] = __expf(sc[n] - m) * inv;
    }
    __syncthreads();

    // z[b][h][c] = sum_n coef * neigh_p[b,n,c] + bias;  2048 outputs, 32/thread.
    for (int j = 0; j < 32; ++j) {
        const int i = t + j * THREADS;
        const int b = i >> 10, h = (i >> 7) & 7, c = i & 127;
        const float* coef = lds_sc + (b * H_HEADS + h) * N_NB;
        float z = 0.f;
        for (int n = 0; n < N_NB; ++n)
            z += coef[n] * lds_p[(b * N_NB + n) * LDS_S + c];
        out[(long)(bBase + b) * (H_HEADS * C_OUT) + h * C_OUT + c] =
            z + bias[h * C_OUT + c];
    }
}

// ---------------------------------------------------------------------------
extern "C" void kernel_launch(void* const* d_in, const int* in_sizes, int n_in,
                              void* d_out, int out_size, void* d_ws, size_t ws_size,
                              hipStream_t stream)
{
    const float* node_feats = (const float*)d_in[0];
    const float* neigh      = (const float*)d_in[1];
    const float* W          = (const float*)d_in[2];
    const float* attnk      = (const float*)d_in[3];
    const float* bias       = (const float*)d_in[4];
    float* out = (float*)d_out;

    float* wa_src = (float*)d_ws;                 // 2048 floats
    float* wt     = wa_src + D_IN * H_HEADS;      // 32768 floats (ws >= ~140 KB)

    gat_prep<<<1, 256, 0, stream>>>(W, attnk, wa_src, wt);
    gat_fused<<<B_TOT / NB_PER_BLK, THREADS, 0, stream>>>(
        node_feats, neigh, attnk, bias, wa_src, wt, out);
}